// Policy_70944269795737
// MI455X (gfx1250) — compile-verified
//
#include <hip/hip_runtime.h>
#include <math.h>

// gfx1250 wave32 WMMA operand vectors
typedef __attribute__((ext_vector_type(2))) float v2f;  // A/B for 16x16x4 f32 (64 f32 / 32 lanes)
typedef __attribute__((ext_vector_type(8))) float v8f;  // C/D 16x16 f32

#define ROWS      4096          // batch B
#define ROWLEN    4096          // D = 2*n floats per row
#define THREADS   256           // 8 waves (wave32)
#define V4_ROW    (ROWLEN / 4)  // 1024 float4 per row
#define V4_THREAD (V4_ROW / THREADS) // 4 float4 per thread

__global__ __launch_bounds__(THREADS)
void policy_xp_logprob_kernel(const float* __restrict__ x,
                              const float* __restrict__ w,
                              float* __restrict__ xp_out,
                              float* __restrict__ lp_out)
{
    const float c = 0.98994949366116653f;   // sqrt(1 - 0.02)
    const float s = 0.14142135623730951f;   // sqrt(0.02)

    const int row = blockIdx.x;
    const int tid = threadIdx.x;

    const float4* x4  = (const float4*)x      + (size_t)row * V4_ROW;
    const float4* w4  = (const float4*)w      + (size_t)row * V4_ROW;
    float4*       xp4 = (float4*)xp_out       + (size_t)row * V4_ROW;
    float4*       lp4 = (float4*)lp_out       + (size_t)row * V4_ROW;

    // ---- streaming pass: xp = c*x + s*w ; accumulate ||xp - (xp - c*x)||^2 ----
    float p = 0.0f;
#pragma unroll
    for (int i = 0; i < V4_THREAD; ++i) {
        const int idx = tid + i * THREADS;          // coalesced float4 access
        const float4 xv = x4[idx];
        const float4 wv = w4[idx];
        float4 xpv;
        xpv.x = fmaf(s, wv.x, c * xv.x);
        xpv.y = fmaf(s, wv.y, c * xv.y);
        xpv.z = fmaf(s, wv.z, c * xv.z);
        xpv.w = fmaf(s, wv.w, c * xv.w);
        xp4[idx] = xpv;
        // reproduce reference rounding: mean = xp - c*x ; diff = xp - mean
        float dx = xpv.x - (xpv.x - c * xv.x);
        float dy = xpv.y - (xpv.y - c * xv.y);
        float dz = xpv.z - (xpv.z - c * xv.z);
        float dw = xpv.w - (xpv.w - c * xv.w);
        p = fmaf(dx, dx, p);
        p = fmaf(dy, dy, p);
        p = fmaf(dz, dz, p);
        p = fmaf(dw, dw, p);
    }

    // ---- block reduction of 256 partials via the matrix pipe ----
    __shared__ float sdata[THREADS];
    sdata[tid] = p;
    __syncthreads();

    const int lane = tid & 31;
    v2f bones; bones.x = 1.0f; bones.y = 1.0f;      // all-ones B: D[m,n] = sum_k A[m,k]
    v8f acc = {};
#pragma unroll
    for (int j = 0; j < 4; ++j) {
        v2f a;
        a.x = sdata[j * 64 + lane];
        a.y = sdata[j * 64 + 32 + lane];
        // v_wmma_f32_16x16x4_f32: accumulate row-sums of each 64-partial tile into C
        acc = __builtin_amdgcn_wmma_f32_16x16x4_f32(
            /*neg_a=*/false, a, /*neg_b=*/false, bones,
            /*c_mod=*/(short)0, acc, /*reuse_a=*/false, /*reuse_b=*/false);
    }
    // lanes 0-15 hold rows 0-7 in VGPRs 0-7; lanes 16-31 hold rows 8-15
    float sc = acc[0] + acc[1] + acc[2] + acc[3] + acc[4] + acc[5] + acc[6] + acc[7];
    float total = sc + __shfl_xor(sc, 16, 32);      // all 256 partials summed (same in every lane)

    // logp = -0.5*total/sigma_var - 0.5*D*log(2*pi*sigma_var)
    const float logp = fmaf(-25.0f, total,           // -0.5/0.02 = -25
                            -0.5f * (float)ROWLEN * logf(2.0f * (float)M_PI * 0.02f));

    // ---- broadcast logp over the row ----
    const float4 lv = make_float4(logp, logp, logp, logp);
#pragma unroll
    for (int i = 0; i < V4_THREAD; ++i) {
        lp4[tid + i * THREADS] = lv;
    }
}

extern "C" void kernel_launch(void* const* d_in, const int* in_sizes, int n_in,
                              void* d_out, int out_size, void* d_ws, size_t ws_size,
                              hipStream_t stream)
{
    const float* x = (const float*)d_in[0];
    const float* w = (const float*)d_in[1];
    float* out = (float*)d_out;
    float* xp  = out;                                  // first output, flat
    float* lp  = out + (size_t)ROWS * ROWLEN;          // second output, flat
    policy_xp_logprob_kernel<<<ROWS, THREADS, 0, stream>>>(x, w, xp, lp);
}